// MoELayer_52544629899956
// MI455X (gfx1250) — compile-verified
//
#include <hip/hip_runtime.h>

#define C_DIM 1024
#define H_DIM 2730
#define E_NUM 8
#define N_TOK 8192

typedef __attribute__((ext_vector_type(2))) float v2f;
typedef __attribute__((ext_vector_type(8))) float v8f;

#define XS 1032   // padded LDS row stride for x tile (16 x 1024)
#define AS 132    // padded LDS row stride for act tile (16 x 128)
#define SMEM_BYTES ((16 * XS + 16 * AS) * 4)

// ---------------------------------------------------------------- router
__global__ __launch_bounds__(256) void router_kernel(
    const float* __restrict__ x, const float* __restrict__ gate_w,
    float* __restrict__ comb, float* __restrict__ partials)
{
    const int tid = threadIdx.x;
    const int n   = blockIdx.x * 256 + tid;

    float logit[E_NUM];
#pragma unroll
    for (int e = 0; e < E_NUM; ++e) logit[e] = 0.f;

    const float* xr = x + (size_t)n * C_DIM;
    for (int c = 0; c < C_DIM; ++c) {
        float xv = xr[c];
        const float* g = gate_w + c * E_NUM;
#pragma unroll
        for (int e = 0; e < E_NUM; ++e) logit[e] += xv * g[e];
    }

    // softmax over 8
    float mx = logit[0];
#pragma unroll
    for (int e = 1; e < E_NUM; ++e) mx = fmaxf(mx, logit[e]);
    float p[E_NUM], s = 0.f;
#pragma unroll
    for (int e = 0; e < E_NUM; ++e) { p[e] = __expf(logit[e] - mx); s += p[e]; }
    float inv = 1.f / s;
#pragma unroll
    for (int e = 0; e < E_NUM; ++e) p[e] *= inv;

    // top-2
    int i0 = 0;
#pragma unroll
    for (int e = 1; e < E_NUM; ++e) if (p[e] > p[i0]) i0 = e;
    int i1 = (i0 == 0) ? 1 : 0;
#pragma unroll
    for (int e = 0; e < E_NUM; ++e) if (e != i0 && p[e] > p[i1]) i1 = e;

    float wsum = p[i0] + p[i1];
    float c0 = p[i0] / wsum, c1 = p[i1] / wsum;
#pragma unroll
    for (int e = 0; e < E_NUM; ++e)
        comb[(size_t)n * E_NUM + e] = (e == i0) ? c0 : ((e == i1) ? c1 : 0.f);

    // deterministic per-block partial sums of probs and counts
    __shared__ float red[256];
    for (int e = 0; e < E_NUM; ++e) {
        red[tid] = p[e];
        __syncthreads();
        for (int off = 128; off > 0; off >>= 1) {
            if (tid < off) red[tid] += red[tid + off];
            __syncthreads();
        }
        if (tid == 0) partials[blockIdx.x * 16 + e] = red[0];
        __syncthreads();
    }
    for (int e = 0; e < E_NUM; ++e) {
        red[tid] = (e == i0 || e == i1) ? 1.f : 0.f;
        __syncthreads();
        for (int off = 128; off > 0; off >>= 1) {
            if (tid < off) red[tid] += red[tid + off];
            __syncthreads();
        }
        if (tid == 0) partials[blockIdx.x * 16 + 8 + e] = red[0];
        __syncthreads();
    }
}

// ---------------------------------------------------------------- aux loss
__global__ void aux_kernel(const float* __restrict__ partials,
                           float* __restrict__ out_aux)
{
    __shared__ float fP[E_NUM], pP[E_NUM];
    int e = threadIdx.x;
    if (e < E_NUM) {
        float sp = 0.f, sc = 0.f;
        for (int b = 0; b < 32; ++b) {
            sp += partials[b * 16 + e];
            sc += partials[b * 16 + 8 + e];
        }
        fP[e] = sc / (float)N_TOK;
        pP[e] = sp / (float)N_TOK;
    }
    __syncthreads();
    if (threadIdx.x == 0) {
        float a = 0.f;
        for (int k = 0; k < E_NUM; ++k) a += fP[k] * pP[k];
        out_aux[0] = (float)E_NUM * a;
    }
}

// ---------------------------------------------------------------- fused MoE FFN
__global__ __launch_bounds__(256) void moe_ffn_kernel(
    const float* __restrict__ x,
    const float* __restrict__ w1, const float* __restrict__ w2,
    const float* __restrict__ w3,
    const float* __restrict__ sw1, const float* __restrict__ sw2,
    const float* __restrict__ sw3,
    const float* __restrict__ comb,
    float* __restrict__ out)
{
    extern __shared__ float smem[];
    float* xs  = smem;             // 16 x XS  : x tile
    float* as_ = smem + 16 * XS;   // 16 x AS  : act tile (16 x 128)

    const int tid   = threadIdx.x;
    const int lane  = tid & 31;
    const int wv    = tid >> 5;          // wave 0..7
    const int t0    = blockIdx.x * 16;   // token base
    const int row16 = lane & 15;
    const int khalf = (lane >> 4) << 1;  // 0 or 2 (K sub-offset per A/B layout)
    const int mhalf = (lane >> 4) << 3;  // 0 or 8 (M offset per C layout)

    // cooperative x-tile load (float4)
    {
        const float4* xg = (const float4*)(x + (size_t)t0 * C_DIM);
        for (int i = tid; i < 16 * (C_DIM / 4); i += 256) {
            int row = i >> 8;          // C_DIM/4 == 256 float4 per row
            int c4  = i & 255;
            float4 v = xg[row * 256 + c4];
            float* d = &xs[row * XS + c4 * 4];
            d[0] = v.x; d[1] = v.y; d[2] = v.z; d[3] = v.w;
        }
    }
    __syncthreads();

    v8f oacc[8];
#pragma unroll
    for (int t = 0; t < 8; ++t)
#pragma unroll
        for (int j = 0; j < 8; ++j) oacc[t][j] = 0.f;

    for (int e = 0; e <= E_NUM; ++e) {
        const float* W1 = (e < E_NUM) ? (w1 + (size_t)e * C_DIM * H_DIM) : sw1;
        const float* W2 = (e < E_NUM) ? (w2 + (size_t)e * C_DIM * H_DIM) : sw2;
        const float* W3 = (e < E_NUM) ? (w3 + (size_t)e * H_DIM * C_DIM) : sw3;

        v8f facc[8];
#pragma unroll
        for (int t = 0; t < 8; ++t)
#pragma unroll
            for (int j = 0; j < 8; ++j) facc[t][j] = 0.f;

        for (int hc = 0; hc < H_DIM; hc += 128) {
            // warm up this chunk's W3 rows (stage-2 B operand) while stage 1 runs
            {
                int pf = hc + lane * 4;
                if (pf < H_DIM)
                    __builtin_prefetch(&W3[(size_t)pf * C_DIM + wv * 128], 0, 1);
            }

            // ---- stage 1: this wave computes act tile cols [hc+16*wv, +16) ----
            const int hbase = hc + wv * 16;
            const int col   = hbase + row16;
            const bool okc  = (col < H_DIM);
            const int colc  = okc ? col : (H_DIM - 1);   // clamp: loads stay legal

            v8f a1, a2;
#pragma unroll
            for (int j = 0; j < 8; ++j) { a1[j] = 0.f; a2[j] = 0.f; }

#pragma unroll 4
            for (int k = 0; k < C_DIM; k += 4) {
                v2f a;
                a.x = xs[row16 * XS + k + khalf];
                a.y = xs[row16 * XS + k + khalf + 1];
                size_t bi = (size_t)(k + khalf) * H_DIM + colc;
                v2f b1, b2;
                b1.x = W1[bi]; b1.y = W1[bi + H_DIM];
                b2.x = W2[bi]; b2.y = W2[bi + H_DIM];
                a1 = __builtin_amdgcn_wmma_f32_16x16x4_f32(
                        false, a, false, b1, (short)0, a1, false, false);
                a2 = __builtin_amdgcn_wmma_f32_16x16x4_f32(
                        false, a, false, b2, (short)0, a2, false, false);
            }

            // silu(h1) * h2 -> LDS act tile (mask padded h-columns to exact 0)
#pragma unroll
            for (int j = 0; j < 8; ++j) {
                int m = j + mhalf;
                float h1 = a1[j];
                float sv = h1 / (1.f + __expf(-h1));
                float val = sv * a2[j];
                as_[m * AS + wv * 16 + row16] = okc ? val : 0.f;
            }
            __syncthreads();

            // ---- stage 2: facc += act(16x128) @ W3[hc:hc+128, wave's 128 cols]
            const int obase = wv * 128;
#pragma unroll 2
            for (int kk = 0; kk < 128; kk += 4) {
                v2f a;
                a.x = as_[row16 * AS + kk + khalf];
                a.y = as_[row16 * AS + kk + khalf + 1];
                int kb  = hc + kk + khalf;
                int kb0 = (kb     < H_DIM) ? kb     : (H_DIM - 1);
                int kb1 = (kb + 1 < H_DIM) ? kb + 1 : (H_DIM - 1);
#pragma unroll
                for (int t = 0; t < 8; ++t) {
                    int ct = obase + t * 16 + row16;
                    v2f b;
                    b.x = W3[(size_t)kb0 * C_DIM + ct];
                    b.y = W3[(size_t)kb1 * C_DIM + ct];
                    facc[t] = __builtin_amdgcn_wmma_f32_16x16x4_f32(
                                  false, a, false, b, (short)0, facc[t], false, false);
                }
            }
            __syncthreads();
        }

        // scale by combine weight (per token row) and accumulate
#pragma unroll
        for (int j = 0; j < 8; ++j) {
            int m = j + mhalf;
            float sc = (e < E_NUM) ? comb[(size_t)(t0 + m) * E_NUM + e] : 1.f;
#pragma unroll
            for (int t = 0; t < 8; ++t) oacc[t][j] += sc * facc[t][j];
        }
    }

    // write 16 x 1024 output tile
#pragma unroll
    for (int t = 0; t < 8; ++t) {
        int cbase = wv * 128 + t * 16 + row16;
#pragma unroll
        for (int j = 0; j < 8; ++j) {
            int m = j + mhalf;
            out[(size_t)(t0 + m) * C_DIM + cbase] = oacc[t][j];
        }
    }
}

// ---------------------------------------------------------------- launch
extern "C" void kernel_launch(void* const* d_in, const int* in_sizes, int n_in,
                              void* d_out, int out_size, void* d_ws, size_t ws_size,
                              hipStream_t stream) {
    const float* x      = (const float*)d_in[0];
    const float* gate_w = (const float*)d_in[1];
    const float* w1     = (const float*)d_in[2];
    const float* w2     = (const float*)d_in[3];
    const float* w3     = (const float*)d_in[4];
    const float* sw1    = (const float*)d_in[5];
    const float* sw2    = (const float*)d_in[6];
    const float* sw3    = (const float*)d_in[7];
    float* out = (float*)d_out;

    float* comb     = (float*)d_ws;                  // N*E floats
    float* partials = comb + (size_t)N_TOK * E_NUM;  // 32*16 floats

    (void)in_sizes; (void)n_in; (void)out_size; (void)ws_size;

    hipFuncSetAttribute((const void*)moe_ffn_kernel,
                        hipFuncAttributeMaxDynamicSharedMemorySize, SMEM_BYTES);

    router_kernel<<<N_TOK / 256, 256, 0, stream>>>(x, gate_w, comb, partials);
    aux_kernel<<<1, 64, 0, stream>>>(partials, out + (size_t)N_TOK * C_DIM);
    moe_ffn_kernel<<<N_TOK / 16, 256, SMEM_BYTES, stream>>>(
        x, w1, w2, w3, sw1, sw2, sw3, comb, out);
}